// AlphaComposition_23768349016287
// MI455X (gfx1250) — compile-verified
//
#include <hip/hip_runtime.h>

namespace {
constexpr int kB = 4, kD = 32, kC = 3, kH = 256, kW = 384;
constexpr int kHW = kH * kW;               // 98304 floats per plane
constexpr int kG4 = kHW / 4;               // 24576 float4 groups per plane
constexpr int kBlk = 256;                  // threads per block (8 waves)
constexpr int kBlocksPerImg = kG4 / kBlk;  // 96
constexpr int kNBuf = 3;                   // triple-buffered pipeline (depth 2)
}  // namespace

// Stage plane (dd) into LDS buffer (bb): alpha + 3 src channels, 16B per lane
// each, via CDNA5 async global->LDS DMA.  Byte offsets:
//   alpha plane stride = kHW*4      = 393216
//   src   plane stride = kC*kHW*4   = 1179648, channel stride = 393216
#define STAGE(dd, bb)                                                          \
  do {                                                                         \
    unsigned aOff = (unsigned)(dd) * 393216u + pixByte;                        \
    unsigned sOff = (unsigned)(dd) * 1179648u + pixByte;                       \
    unsigned l0 = ldsBase + ((unsigned)(bb) * 4u * kBlk + (unsigned)tid) * 16u;\
    unsigned l1 = l0 + kBlk * 16u;                                             \
    unsigned l2 = l1 + kBlk * 16u;                                             \
    unsigned l3 = l2 + kBlk * 16u;                                             \
    asm volatile(                                                              \
        "s_wait_dscnt 0x0\n\t"                                                 \
        "global_load_async_to_lds_b128 %0, %4, %6\n\t"                         \
        "global_load_async_to_lds_b128 %1, %5, %7\n\t"                         \
        "global_load_async_to_lds_b128 %2, %5, %7 offset:393216\n\t"           \
        "global_load_async_to_lds_b128 %3, %5, %7 offset:786432"               \
        :: "v"(l0), "v"(l1), "v"(l2), "v"(l3), "v"(aOff), "v"(sOff),           \
           "s"(aBase64), "s"(sBase64)                                          \
        : "memory");                                                           \
  } while (0)

__global__ __launch_bounds__(kBlk)
void AlphaComposition_kernel(const float* __restrict__ src,
                             const float* __restrict__ alpha,
                             float* __restrict__ out) {
  // [buf 0..2][region: alpha,c0,c1,c2][tid] of float4  -> 48 KB
  __shared__ float4 smem[kNBuf * 4 * kBlk];

  const int tid = threadIdx.x;
  const int b   = blockIdx.x / kBlocksPerImg;               // image (uniform)
  const int g0  = (blockIdx.x % kBlocksPerImg) * kBlk + tid; // float4 group in plane
  const unsigned pixByte = (unsigned)g0 * 16u;

  const unsigned long long aBase64 =
      (unsigned long long)(const void*)(alpha + (size_t)b * kD * kHW);
  const unsigned long long sBase64 =
      (unsigned long long)(const void*)(src + (size_t)b * kD * kC * kHW);
  // low 32 bits of a generic pointer to LDS == LDS byte offset
  const unsigned ldsBase = (unsigned)(size_t)(&smem[0]);

  float resist[4] = {1.f, 1.f, 1.f, 1.f};
  float vsum[4]   = {0.f, 0.f, 0.f, 0.f};
  float acc0[4]   = {0.f, 0.f, 0.f, 0.f};
  float acc1[4]   = {0.f, 0.f, 0.f, 0.f};
  float acc2[4]   = {0.f, 0.f, 0.f, 0.f};

  // Prime the pipeline: planes 0 and 1 in flight (8 async ops).
  STAGE(0, 0);
  STAGE(1, 1);

  int buf = 0;
  for (int d = 0; d < kD; ++d) {
    if (d + 2 < kD) {
      int nb = buf + 2; if (nb >= kNBuf) nb -= kNBuf;
      STAGE(d + 2, nb);                                  // 12 ops in flight
      asm volatile("s_wait_asynccnt 0x8" ::: "memory");  // stage d landed
    } else if (d + 1 < kD) {
      asm volatile("s_wait_asynccnt 0x4" ::: "memory");
    } else {
      asm volatile("s_wait_asynccnt 0x0" ::: "memory");
    }

    const int rb = buf * 4 * kBlk + tid;
    float4 a4 = smem[rb];
    float4 c0 = smem[rb + kBlk];
    float4 c1 = smem[rb + 2 * kBlk];
    float4 c2 = smem[rb + 3 * kBlk];

    float av[4]  = {a4.x, a4.y, a4.z, a4.w};
    float s0v[4] = {c0.x, c0.y, c0.z, c0.w};
    float s1v[4] = {c1.x, c1.y, c1.z, c1.w};
    float s2v[4] = {c2.x, c2.y, c2.z, c2.w};
#pragma unroll
    for (int j = 0; j < 4; ++j) {
      float a = av[j];
      float v = resist[j] * a;          // visibility (unnormalized)
      vsum[j] += v;
      acc0[j] = fmaf(s0v[j], v, acc0[j]);
      acc1[j] = fmaf(s1v[j], v, acc1[j]);
      acc2[j] = fmaf(s2v[j], v, acc2[j]);
      resist[j] = resist[j] * (1.0f - a);  // exclusive cumprod of (1-alpha)
    }
    buf = buf + 1; if (buf >= kNBuf) buf = 0;
  }

  float inv[4];
#pragma unroll
  for (int j = 0; j < 4; ++j) inv[j] = 1.0f / fmaxf(vsum[j], 1e-7f);

  float4 o0 = make_float4(acc0[0] * inv[0], acc0[1] * inv[1],
                          acc0[2] * inv[2], acc0[3] * inv[3]);
  float4 o1 = make_float4(acc1[0] * inv[0], acc1[1] * inv[1],
                          acc1[2] * inv[2], acc1[3] * inv[3]);
  float4 o2 = make_float4(acc2[0] * inv[0], acc2[1] * inv[1],
                          acc2[2] * inv[2], acc2[3] * inv[3]);

  float4* outv = reinterpret_cast<float4*>(out) + (size_t)b * kC * kG4 + g0;
  outv[0]        = o0;
  outv[kG4]      = o1;
  outv[2 * kG4]  = o2;
}

extern "C" void kernel_launch(void* const* d_in, const int* in_sizes, int n_in,
                              void* d_out, int out_size, void* d_ws, size_t ws_size,
                              hipStream_t stream) {
  (void)in_sizes; (void)n_in; (void)out_size; (void)d_ws; (void)ws_size;
  const float* src   = (const float*)d_in[0];  // [B,D,C,H,W] fp32
  const float* alpha = (const float*)d_in[1];  // [B,D,1,H,W] fp32
  float* out = (float*)d_out;                  // [B,C,H,W]  fp32

  dim3 grid(kB * kBlocksPerImg);  // 384 blocks, exact cover
  dim3 block(kBlk);
  AlphaComposition_kernel<<<grid, block, 0, stream>>>(src, alpha, out);
}